// SlidingWindowAttention_9835475107985
// MI455X (gfx1250) — compile-verified
//
#include <hip/hip_runtime.h>
#include <math.h>

typedef __attribute__((ext_vector_type(2))) float v2f;
typedef __attribute__((ext_vector_type(4))) float v4f;
typedef __attribute__((ext_vector_type(8))) float v8f;

#define B_        2
#define H_        16
#define L_        2048
#define D_        64
#define HALF_WIN  128
#define WCOLS     272            // 16 queries + 2*128 window
#define KTILES    17             // 272 / 16
#define LDS_STRIDE 276           // mult-of-4 (16B rows) & conflict-free striping
#define WPB       8              // waves per block
#define WAVE_LDS  (16 * LDS_STRIDE + 16)   // score band + invSum[16]
#define BIAS_N    287                       // diff in [-143, 143]
#define SMEM_FLOATS (WPB * WAVE_LDS + BIAS_N + 1)
#define SMEM_BYTES  (SMEM_FLOATS * sizeof(float))

// ---------------------------------------------------------------------------
// Banded sliding-window attention, one wave32 per 16-query tile.
// Writes the FULL attention rows (zeros outside the band) in a single pass
// with b128 non-temporal stores -> 512 MB once instead of 512+67 MB twice.
// ---------------------------------------------------------------------------
__global__ __launch_bounds__(256) void swa_main(
    const float* __restrict__ Q, const float* __restrict__ K,
    const float* __restrict__ V, const float* __restrict__ mask,
    const float* __restrict__ scal, const float* __restrict__ taup,
    float* __restrict__ out, float* __restrict__ attn)
{
    extern __shared__ float smem[];
    float* biasTab = smem + WPB * WAVE_LDS;     // [BIAS_N] exp bias lookup

    const int tid  = threadIdx.x;
    const int wave = tid >> 5;
    const int lane = tid & 31;
    const int row  = lane & 15;      // M (or N / K-row) index within tile
    const int half = lane >> 4;      // 0: lanes 0-15, 1: lanes 16-31

    // ---- block-level bias table: bias(diff) = exp(-|diff/sc|/tau) ---------
    {
        const float sc   = scal[0];
        const float tauv = taup[0];
        for (int i = tid; i < BIAS_N; i += 256) {
            const float d = (float)(i - 143);
            biasTab[i] = expf(-fabsf(d / sc) / tauv);   // exact, done once
        }
    }
    __syncthreads();

    const int tile = blockIdx.x * WPB + wave;   // 0 .. 4095
    const int bh   = tile >> 7;                 // / (L_/16) = 128 q-tiles
    const int qt   = tile & 127;
    const int q0   = qt << 4;
    const int b    = bh >> 4;                   // H_ = 16

    float* sb  = smem + wave * WAVE_LDS;        // [16][LDS_STRIDE]
    float* inv = sb + 16 * LDS_STRIDE;          // [16] 1/rowsum

    const size_t bhLD = (size_t)bh * L_ * D_;
    const float* Qb = Q + bhLD + (size_t)q0 * D_;
    const float* Kb = K + bhLD;
    const float* Vb = V + bhLD;

    // ---- row-constant mask bits, hoisted out of the tile loop -------------
    bool mq[8];
    #pragma unroll
    for (int j = 0; j < 8; ++j)
        mq[j] = mask[(size_t)b * L_ + q0 + j + (half << 3)] != 0.0f;

    // ---- load all Q A-fragments: 16 steps of K-dim 4 over d=64 ------------
    v2f qa[16];
    {
        const v2f* qp2 = (const v2f*)(Qb + row * D_);
        #pragma unroll
        for (int kk = 0; kk < 16; ++kk) qa[kk] = qp2[2 * kk + half];
    }

    // ---- scores: 17 tiles of S = Q * K^T, masked + biased, into LDS -------
    for (int t = 0; t < KTILES; ++t) {
        const int k0 = q0 - HALF_WIN + (t << 4);
        const int kg = k0 + row;                       // this lane's K row / N col
        const int kc = kg < 0 ? 0 : (kg >= L_ ? L_ - 1 : kg);
        const v2f* kp2 = (const v2f*)(Kb + (size_t)kc * D_);

        v8f c = {};
        #pragma unroll
        for (int kk = 0; kk < 16; ++kk) {
            v2f bf = kp2[2 * kk + half];
            c = __builtin_amdgcn_wmma_f32_16x16x4_f32(
                    false, qa[kk], false, bf, (short)0, c, false, false);
        }

        const bool kin = (kg >= 0) && (kg < L_);
        const int cb = (t << 4) + row;                 // band column 0..271
        // diff = (j + 8*half) + 128 - cb;  table index = diff + 143
        const int tabBase = (half << 3) + 128 - cb + 143;
        #pragma unroll
        for (int j = 0; j < 8; ++j) {
            const int M = j + (half << 3);
            const int diff = M + HALF_WIN - cb;
            const float bias = biasTab[tabBase + j];
            const bool inw = (diff <= HALF_WIN) && (diff >= -HALF_WIN);
            const float s  = c[j] * 0.125f;            // 1/sqrt(64)
            const float base = (inw && mq[j]) ? s : -1e9f;
            sb[M * LDS_STRIDE + cb] = kin ? base - bias : -INFINITY;
        }
    }

    // ---- softmax over the 272-wide band: 2 lanes per row ------------------
    {
        float* prow = sb + row * LDS_STRIDE;
        float m = -INFINITY;
        for (int cix = half; cix < WCOLS; cix += 2) m = fmaxf(m, prow[cix]);
        m = fmaxf(m, __shfl_xor(m, 16, 32));
        float ssum = 0.f;
        for (int cix = half; cix < WCOLS; cix += 2) {
            float e = __expf(prow[cix] - m);
            prow[cix] = e;                 // store unnormalized p
            ssum += e;
        }
        ssum += __shfl_xor(ssum, 16, 32);
        if (half == 0) inv[row] = 1.0f / ssum;
    }

    // ---- write FULL attention rows: zeros outside band, p*inv inside ------
    // band start (q0-128) is 16-float aligned; 272 = 68 float4 groups.
    {
        const int g0 = (q0 - HALF_WIN) >> 2;           // first band group (may be <0)
        for (int r = 0; r < 16; ++r) {
            const float irs = inv[r];
            v4f* arow4 = (v4f*)(attn + ((size_t)bh * L_ + (q0 + r)) * L_);
            const v4f* prow4 = (const v4f*)(sb + r * LDS_STRIDE);
            for (int g = lane; g < (L_ / 4); g += 32) {
                v4f val = {0.f, 0.f, 0.f, 0.f};
                const int rel = g - g0;
                if ((unsigned)rel < (unsigned)(WCOLS / 4))
                    val = prow4[rel] * irs;
                __builtin_nontemporal_store(val, &arow4[g]);
            }
        }
    }

    // ---- O = P * V : 4 output n-tiles, 68 K-steps of 4 --------------------
    for (int t4 = 0; t4 < 4; ++t4) {
        const int n0 = t4 << 4;
        v8f c = {};
        #pragma unroll 4
        for (int kk = 0; kk < 68; ++kk) {
            const int kd = (kk << 2) + (half << 1);
            v2f a;
            a.x = sb[row * LDS_STRIDE + kd];
            a.y = sb[row * LDS_STRIDE + kd + 1];
            const int k1 = q0 - HALF_WIN + kd;
            const int k2 = k1 + 1;
            const int k1c = k1 < 0 ? 0 : (k1 >= L_ ? L_ - 1 : k1);
            const int k2c = k2 < 0 ? 0 : (k2 >= L_ ? L_ - 1 : k2);
            v2f bf;                       // p==0 exactly for clamped rows
            bf.x = Vb[(size_t)k1c * D_ + n0 + row];
            bf.y = Vb[(size_t)k2c * D_ + n0 + row];
            c = __builtin_amdgcn_wmma_f32_16x16x4_f32(
                    false, a, false, bf, (short)0, c, false, false);
        }
        #pragma unroll
        for (int j = 0; j < 8; ++j) {
            const int M = j + (half << 3);
            out[bhLD + (size_t)(q0 + M) * D_ + n0 + row] = c[j] * inv[M];
        }
    }
}

// ---------------------------------------------------------------------------
extern "C" void kernel_launch(void* const* d_in, const int* in_sizes, int n_in,
                              void* d_out, int out_size, void* d_ws, size_t ws_size,
                              hipStream_t stream) {
    (void)in_sizes; (void)n_in; (void)out_size; (void)d_ws; (void)ws_size;
    const float* Q    = (const float*)d_in[0];
    const float* K    = (const float*)d_in[1];
    const float* V    = (const float*)d_in[2];
    const float* mask = (const float*)d_in[3];
    const float* scal = (const float*)d_in[4];
    const float* tau  = (const float*)d_in[5];

    float* out  = (float*)d_out;
    float* attn = out + (size_t)B_ * H_ * L_ * D_;

    // banded WMMA attention: 4096 wave-tiles, 8 waves per block.
    // Each wave writes its 16 full attention rows (single pass over 512 MB).
    const int tiles  = B_ * H_ * (L_ / 16);
    const int blocks = tiles / WPB;                 // 512
    swa_main<<<blocks, 256, SMEM_BYTES, stream>>>(Q, K, V, mask, scal, tau,
                                                  out, attn);
}